// RtNet25DDualRENet_72791105732657
// MI455X (gfx1250) — compile-verified
//
#include <hip/hip_runtime.h>
#include <math.h>

// ---------------------------------------------------------------------------
// RENet-style dual-branch network on gfx1250 (MI455X), fp32 end-to-end.
// All GEMM-shaped ops use V_WMMA_F32_16X16X4_F32 (full fp32 precision).
// Branchless masked operand fetch keeps EXEC all-ones through the WMMA loops.
// ---------------------------------------------------------------------------

#define BB   64            // batch
#define CC   960           // encoder dim
#define HH   7
#define HWP  49            // 7*7
#define CO   96            // scr mid channels
#define UV   25            // 5*5 window
#define BN_INV 0.9999950000374997f   // 1/sqrt(1+1e-5)

typedef float v2f __attribute__((ext_vector_type(2)));
typedef float v8f __attribute__((ext_vector_type(8)));

// One 16x16x4 fp32 WMMA step. A fragment: lane L holds A[M=L&15][kbase + 2*(L>>4) + {0,1}],
// B fragment: lane L holds B[kbase + 2*(L>>4) + {0,1}][N=L&15]. D/C: VGPR j = row j + 8*(L>>4).
__device__ __forceinline__ v8f wmma4(v8f acc, float a0, float a1, float b0, float b1) {
    v2f a; a.x = a0; a.y = a1;
    v2f b; b.x = b0; b.y = b1;
    return __builtin_amdgcn_wmma_f32_16x16x4_f32(false, a, false, b, (short)0, acc, false, false);
}

__device__ __forceinline__ float block_reduce_sum(float v, float* red) {
    int t = threadIdx.x;
    red[t] = v; __syncthreads();
    for (int s = blockDim.x >> 1; s > 0; s >>= 1) {
        if (t < s) red[t] += red[t + s];
        __syncthreads();
    }
    float r = red[0];
    __syncthreads();
    return r;
}

// --------------------------- branch: prep (relu + l2norm over C) -----------
__global__ void k_prep(const float* __restrict__ x, float* __restrict__ xn) {
    __shared__ float red[128];
    int bp = blockIdx.x;
    int b = bp / HWP, p = bp % HWP;
    const float* xb = x + (size_t)b * CC * HWP + p;
    float ss = 0.f;
    for (int c = threadIdx.x; c < CC; c += 128) {
        float v = xb[(size_t)c * HWP]; v = v > 0.f ? v : 0.f; ss += v * v;
    }
    float tot = block_reduce_sum(ss, red);
    float inv = 1.f / fmaxf(sqrtf(tot), 1e-12f);
    float* ob = xn + (size_t)b * CC * HWP + p;
    for (int c = threadIdx.x; c < CC; c += 128) {
        float v = xb[(size_t)c * HWP]; v = v > 0.f ? v : 0.f;
        ob[(size_t)c * HWP] = v * inv;
    }
}

// ----- GEMM1: out[b,o,p,uv] = relu(bn( sum_c w_in[o,c] * xn[c,p]*xn[c,shift(p,uv)] )) -----
// grid: (6 o-tiles, BB*UV); block 128 = 4 waves (p-tiles). Branchless masked B fetch.
__global__ void k_gemm1(const float* __restrict__ xn, const float* __restrict__ w_in,
                        const float* __restrict__ bnp, float* __restrict__ g1) {
    int wave = threadIdx.x >> 5, lane = threadIdx.x & 31;
    int ot = blockIdx.x;
    int bz = blockIdx.y; int b = bz / UV, uv = bz % UV;
    int du = uv / 5 - 2, dv = uv % 5 - 2;
    int m  = lane & 15;
    int kk = (lane >> 4) << 1;
    int arow = ot * 16 + m;           // o index for A loads
    int n    = wave * 16 + m;         // column p for B loads
    int h = n / HH, w = n % HH;
    int ph = h + du, pw = w + dv;
    bool svalid = (n < HWP) && ((unsigned)ph < HH) && ((unsigned)pw < HH);
    float mask = svalid ? 1.f : 0.f;
    const float* xb    = xn + (size_t)b * CC * HWP;
    const float* xcol  = xb + (n < HWP ? n : 0);                 // clamped, always in-bounds
    const float* xcols = xb + (svalid ? ph * HH + pw : 0);       // clamped, always in-bounds
    const float* arow_p = w_in + (size_t)arow * CC;
    v8f acc = {};
    for (int k0 = 0; k0 < CC; k0 += 64) {
        __builtin_prefetch(arow_p + k0 + 64, 0, 3);
        for (int kq = 0; kq < 64; kq += 4) {
            int k = k0 + kq + kk;
            float a0 = arow_p[k];
            float a1 = arow_p[k + 1];
            float b0 = mask * xcol[(size_t)k * HWP]       * xcols[(size_t)k * HWP];
            float b1 = mask * xcol[(size_t)(k + 1) * HWP] * xcols[(size_t)(k + 1) * HWP];
            acc = wmma4(acc, a0, a1, b0, b1);
        }
    }
    int col = wave * 16 + (lane & 15);
    if (col < HWP) {
        int rbase = ot * 16 + ((lane >> 4) << 3);
        for (int j = 0; j < 8; ++j) {
            int r = rbase + j;
            float v = acc[j] * BN_INV * bnp[r] + bnp[CO + r];
            v = v > 0.f ? v : 0.f;
            g1[(((size_t)b * CO + r) * HWP + col) * UV + uv] = v;
        }
    }
}

// ----- conv1 (1,3,3) over (u,v): g1[b,96,49,5,5] -> t2[b,96,49,3,3], relu(bn) -----
__global__ void k_conv1(const float* __restrict__ g1, const float* __restrict__ w1,
                        const float* __restrict__ bnp, float* __restrict__ t2) {
    int idx = blockIdx.x * blockDim.x + threadIdx.x;
    const int total = BB * CO * HWP * 9;
    if (idx >= total) return;
    int v2 = idx % 3; int tmp = idx / 3;
    int u2 = tmp % 3; tmp /= 3;
    int p  = tmp % HWP; tmp /= HWP;
    int o  = tmp % CO;  int b = tmp / CO;
    const float* xb = g1 + (size_t)b * CO * HWP * UV + (size_t)p * UV;
    float s = 0.f;
    for (int ci = 0; ci < CO; ++ci) {
        const float* xc = xb + (size_t)ci * HWP * UV;
        const float* wc = w1 + ((size_t)o * CO + ci) * 9;
        for (int du = 0; du < 3; ++du)
            for (int dv = 0; dv < 3; ++dv)
                s += wc[du * 3 + dv] * xc[(u2 + du) * 5 + (v2 + dv)];
    }
    s = s * BN_INV * bnp[o] + bnp[CO + o];
    t2[(((size_t)b * CO + o) * HWP + p) * 9 + u2 * 3 + v2] = s > 0.f ? s : 0.f;
}

// ----- conv2 (1,3,3): t2[b,96,49,3,3] -> t3[b,96,49], relu(bn) -----
__global__ void k_conv2(const float* __restrict__ t2, const float* __restrict__ w2,
                        const float* __restrict__ bnp, float* __restrict__ t3) {
    int idx = blockIdx.x * blockDim.x + threadIdx.x;
    const int total = BB * CO * HWP;
    if (idx >= total) return;
    int p = idx % HWP; int tmp = idx / HWP;
    int o = tmp % CO;  int b = tmp / CO;
    float s = 0.f;
    for (int ci = 0; ci < CO; ++ci) {
        const float* xc = t2 + (((size_t)b * CO + ci) * HWP + p) * 9;
        const float* wc = w2 + ((size_t)o * CO + ci) * 9;
        for (int t = 0; t < 9; ++t) s += wc[t] * xc[t];
    }
    s = s * BN_INV * bnp[o] + bnp[CO + o];
    t3[((size_t)b * CO + o) * HWP + p] = s > 0.f ? s : 0.f;
}

// ----- w_out GEMM + bn_out + residual + relu: dst[b,960,49] -----
// grid (60 m-tiles, BB); block 128 = 4 waves (p-tiles)
__global__ void k_gemm_out(const float* __restrict__ t3, const float* __restrict__ w_out,
                           const float* __restrict__ bnp, const float* __restrict__ xorig,
                           float* __restrict__ dst) {
    int wave = threadIdx.x >> 5, lane = threadIdx.x & 31;
    int mt = blockIdx.x, b = blockIdx.y;
    int m = lane & 15, kk = (lane >> 4) << 1;
    int arow = mt * 16 + m;
    int n = wave * 16 + m;
    float maskB = (n < HWP) ? 1.f : 0.f;
    int nc = (n < HWP) ? n : 0;
    const float* Bb = t3 + (size_t)b * CO * HWP + nc;
    const float* Ar = w_out + (size_t)arow * CO;
    v8f acc = {};
    for (int k0 = 0; k0 < CO; k0 += 4) {
        int k = k0 + kk;
        float a0 = Ar[k];
        float a1 = Ar[k + 1];
        float b0 = maskB * Bb[(size_t)k * HWP];
        float b1 = maskB * Bb[(size_t)(k + 1) * HWP];
        acc = wmma4(acc, a0, a1, b0, b1);
    }
    int col = wave * 16 + (lane & 15);
    if (col < HWP) {
        int rbase = mt * 16 + ((lane >> 4) << 3);
        for (int j = 0; j < 8; ++j) {
            int r = rbase + j;
            size_t off = ((size_t)b * CC + r) * HWP + col;
            float v = acc[j] * BN_INV * bnp[r] + bnp[CC + r] + xorig[off];
            dst[off] = v > 0.f ? v : 0.f;
        }
    }
}

// ----- subtract channel mean in place: x[b,960,49] -----
__global__ void k_meansub(float* __restrict__ x) {
    __shared__ float red[128];
    int bp = blockIdx.x;
    int b = bp / HWP, p = bp % HWP;
    float* xb = x + (size_t)b * CC * HWP + p;
    float s = 0.f;
    for (int c = threadIdx.x; c < CC; c += 128) s += xb[(size_t)c * HWP];
    float mean = block_reduce_sum(s, red) / (float)CC;
    for (int c = threadIdx.x; c < CC; c += 128) xb[(size_t)c * HWP] -= mean;
}

// ----- cca1x1 GEMM: z[b,96,49] = relu(bn( w[96,960] * xpn[b,960,49] )) -----
// grid (6 m-tiles, BB); block 128 = 4 waves (p-tiles)
__global__ void k_gemm_cca(const float* __restrict__ xpn, const float* __restrict__ wcca,
                           const float* __restrict__ bnp, float* __restrict__ z) {
    int wave = threadIdx.x >> 5, lane = threadIdx.x & 31;
    int ot = blockIdx.x, b = blockIdx.y;
    int m = lane & 15, kk = (lane >> 4) << 1;
    int arow = ot * 16 + m;
    int n = wave * 16 + m;
    float maskB = (n < HWP) ? 1.f : 0.f;
    int nc = (n < HWP) ? n : 0;
    const float* Bb = xpn + (size_t)b * CC * HWP + nc;
    const float* Ar = wcca + (size_t)arow * CC;
    v8f acc = {};
    for (int k0 = 0; k0 < CC; k0 += 64) {
        __builtin_prefetch(Ar + k0 + 64, 0, 3);
        for (int kq = 0; kq < 64; kq += 4) {
            int k = k0 + kq + kk;
            float a0 = Ar[k];
            float a1 = Ar[k + 1];
            float b0 = maskB * Bb[(size_t)k * HWP];
            float b1 = maskB * Bb[(size_t)(k + 1) * HWP];
            acc = wmma4(acc, a0, a1, b0, b1);
        }
    }
    int col = wave * 16 + (lane & 15);
    if (col < HWP) {
        int rbase = ot * 16 + ((lane >> 4) << 3);
        for (int j = 0; j < 8; ++j) {
            int r = rbase + j;
            float v = acc[j] * BN_INV * bnp[r] + bnp[CO + r];
            z[((size_t)b * CO + r) * HWP + col] = v > 0.f ? v : 0.f;
        }
    }
}

// ----- l2 normalize over 96 channels in place (eps 1e-8) -----
__global__ void k_l2n96(float* __restrict__ z) {
    __shared__ float red[128];
    int bp = blockIdx.x;
    int b = bp / HWP, p = bp % HWP;
    float* zb = z + (size_t)b * CO * HWP + p;
    float ss = 0.f;
    for (int c = threadIdx.x; c < CO; c += 128) { float v = zb[(size_t)c * HWP]; ss += v * v; }
    float tot = block_reduce_sum(ss, red);
    float inv = 1.f / fmaxf(sqrtf(tot), 1e-8f);
    for (int c = threadIdx.x; c < CO; c += 128) zb[(size_t)c * HWP] *= inv;
}

// ----- corr[b,i,j] = sum_c spt[b,c,i]*qry[b,c,j]. grid (4 i-tiles, BB); 4 waves = j-tiles
__global__ void k_gemm_corr(const float* __restrict__ spt, const float* __restrict__ qry,
                            float* __restrict__ corr) {
    int wave = threadIdx.x >> 5, lane = threadIdx.x & 31;
    int it = blockIdx.x, b = blockIdx.y;
    int m = lane & 15, kk = (lane >> 4) << 1;
    int i = it * 16 + m;
    int n = wave * 16 + m;
    float maskA = (i < HWP) ? 1.f : 0.f;
    float maskB = (n < HWP) ? 1.f : 0.f;
    const float* Sb = spt + (size_t)b * CO * HWP + ((i < HWP) ? i : 0);
    const float* Qb = qry + (size_t)b * CO * HWP + ((n < HWP) ? n : 0);
    v8f acc = {};
    for (int k0 = 0; k0 < CO; k0 += 4) {
        int k = k0 + kk;
        float a0 = maskA * Sb[(size_t)k * HWP];
        float a1 = maskA * Sb[(size_t)(k + 1) * HWP];
        float b0 = maskB * Qb[(size_t)k * HWP];
        float b1 = maskB * Qb[(size_t)(k + 1) * HWP];
        acc = wmma4(acc, a0, a1, b0, b1);
    }
    int col = wave * 16 + (lane & 15);
    if (col < HWP) {
        int rbase = it * 16 + ((lane >> 4) << 3);
        for (int j = 0; j < 8; ++j) {
            int r = rbase + j;
            if (r < HWP) corr[((size_t)b * HWP + r) * HWP + col] = acc[j];
        }
    }
}

// ----- SepConv4d #1 (c = 1) -----
__global__ void k_sep1a(const float* __restrict__ corr, const float* __restrict__ w,
                        const float* __restrict__ bnp, float* __restrict__ ta) {
    int idx = blockIdx.x * blockDim.x + threadIdx.x;
    if (idx >= BB * HWP * HWP) return;
    int j = idx % HWP; int tmp = idx / HWP;
    int is = tmp % HWP; int b = tmp / HWP;
    int hs = is / HH, ws = is % HH;
    float s = 0.f;
    for (int du = -1; du <= 1; ++du)
        for (int dv = -1; dv <= 1; ++dv) {
            int hh = hs + du, ww = ws + dv;
            if ((unsigned)hh < HH && (unsigned)ww < HH)
                s += w[(du + 1) * 3 + (dv + 1)] * corr[((size_t)b * HWP + hh * HH + ww) * HWP + j];
        }
    s = s * BN_INV * bnp[0] + bnp[1];
    ta[idx] = s > 0.f ? s : 0.f;
}

__global__ void k_sep1b(const float* __restrict__ ta, const float* __restrict__ w,
                        const float* __restrict__ bnp, float* __restrict__ tb) {
    int idx = blockIdx.x * blockDim.x + threadIdx.x;
    if (idx >= BB * HWP * HWP) return;
    int j = idx % HWP; int tmp = idx / HWP;
    int is = tmp % HWP; int b = tmp / HWP;
    int hq = j / HH, wq = j % HH;
    float s = 0.f;
    for (int du = -1; du <= 1; ++du)
        for (int dv = -1; dv <= 1; ++dv) {
            int hh = hq + du, ww = wq + dv;
            if ((unsigned)hh < HH && (unsigned)ww < HH)
                s += w[(du + 1) * 3 + (dv + 1)] * ta[((size_t)b * HWP + is) * HWP + hh * HH + ww];
        }
    tb[idx] = s * BN_INV * bnp[0] + bnp[1];
}

__global__ void k_sep1c(const float* __restrict__ tb, const float* __restrict__ pw,
                        const float* __restrict__ bnp, float* __restrict__ c4a) {
    int idx = blockIdx.x * blockDim.x + threadIdx.x;
    if (idx >= BB * 16 * HWP * HWP) return;
    int sp = idx % (HWP * HWP); int tmp = idx / (HWP * HWP);
    int o = tmp % 16; int b = tmp / 16;
    float s = pw[o] * tb[(size_t)b * HWP * HWP + sp];
    s = s * BN_INV * bnp[o] + bnp[16 + o];
    c4a[idx] = s > 0.f ? s : 0.f;            // relu after first SepConv4d
}

// ----- SepConv4d #2 (c = 16) -----
__global__ void k_sep2a(const float* __restrict__ c4a, const float* __restrict__ w,
                        const float* __restrict__ bnp, float* __restrict__ s1) {
    int idx = blockIdx.x * blockDim.x + threadIdx.x;
    if (idx >= BB * 16 * HWP * HWP) return;
    int j = idx % HWP; int tmp = idx / HWP;
    int is = tmp % HWP; tmp /= HWP;
    int o = tmp % 16; int b = tmp / 16;
    int hs = is / HH, ws = is % HH;
    float s = 0.f;
    for (int c = 0; c < 16; ++c) {
        const float* wc = w + ((size_t)o * 16 + c) * 9;
        const float* xc = c4a + ((size_t)b * 16 + c) * HWP * HWP;
        for (int du = -1; du <= 1; ++du)
            for (int dv = -1; dv <= 1; ++dv) {
                int hh = hs + du, ww = ws + dv;
                if ((unsigned)hh < HH && (unsigned)ww < HH)
                    s += wc[(du + 1) * 3 + (dv + 1)] * xc[(size_t)(hh * HH + ww) * HWP + j];
            }
    }
    s = s * BN_INV * bnp[o] + bnp[16 + o];
    s1[idx] = s > 0.f ? s : 0.f;
}

__global__ void k_sep2b(const float* __restrict__ s1, const float* __restrict__ w,
                        const float* __restrict__ bnp, float* __restrict__ s2) {
    int idx = blockIdx.x * blockDim.x + threadIdx.x;
    if (idx >= BB * 16 * HWP * HWP) return;
    int j = idx % HWP; int tmp = idx / HWP;
    int is = tmp % HWP; tmp /= HWP;
    int o = tmp % 16; int b = tmp / 16;
    int hq = j / HH, wq = j % HH;
    float s = 0.f;
    for (int c = 0; c < 16; ++c) {
        const float* wc = w + ((size_t)o * 16 + c) * 9;
        const float* xc = s1 + (((size_t)b * 16 + c) * HWP + is) * HWP;
        for (int du = -1; du <= 1; ++du)
            for (int dv = -1; dv <= 1; ++dv) {
                int hh = hq + du, ww = wq + dv;
                if ((unsigned)hh < HH && (unsigned)ww < HH)
                    s += wc[(du + 1) * 3 + (dv + 1)] * xc[hh * HH + ww];
            }
    }
    s2[idx] = s * BN_INV * bnp[o] + bnp[16 + o];
}

__global__ void k_sep2c(const float* __restrict__ s2, const float* __restrict__ pw,
                        const float* __restrict__ bnp, float* __restrict__ c4b) {
    int idx = blockIdx.x * blockDim.x + threadIdx.x;
    if (idx >= BB * HWP * HWP) return;
    int sp = idx % (HWP * HWP); int b = idx / (HWP * HWP);
    float s = 0.f;
    for (int c = 0; c < 16; ++c)
        s += pw[c] * s2[((size_t)b * 16 + c) * HWP * HWP + sp];
    c4b[idx] = s * BN_INV * bnp[0] + bnp[1];   // no relu after second SepConv4d
}

// ----- gauss + softmax + attention sums. one block per batch, 64 threads -----
__global__ void k_attn(const float* __restrict__ c4b, float* __restrict__ attn_s,
                       float* __restrict__ attn_q) {
    __shared__ float sm[HWP * HWP];
    int b = blockIdx.x, t = threadIdx.x;
    const float* cb = c4b + (size_t)b * HWP * HWP;
    // columns: gauss+softmax over spt index i (axis=1), fixed qry j=t
    if (t < HWP) {
        float v[HWP];
        float mean = 0.f;
        for (int i = 0; i < HWP; ++i) { v[i] = cb[(size_t)i * HWP + t]; mean += v[i]; }
        mean /= (float)HWP;
        float var = 0.f;
        for (int i = 0; i < HWP; ++i) { float d = v[i] - mean; var += d * d; }
        var /= (float)(HWP - 1);
        float inv = 1.f / sqrtf(var + 1e-5f);
        float mx = -3.4e38f;
        for (int i = 0; i < HWP; ++i) { v[i] = (v[i] - mean) * inv; mx = fmaxf(mx, v[i]); }
        float se = 0.f;
        for (int i = 0; i < HWP; ++i) { v[i] = expf(v[i] - mx); se += v[i]; }
        float rs = 1.f / se;
        for (int i = 0; i < HWP; ++i) sm[i * HWP + t] = v[i] * rs;
    }
    __syncthreads();
    if (t < HWP) {
        float s = 0.f;
        for (int j = 0; j < HWP; ++j) s += sm[t * HWP + j];
        attn_s[b * HWP + t] = s;
    }
    __syncthreads();
    // rows: gauss+softmax over qry index j (axis=3), fixed spt i=t
    if (t < HWP) {
        float v[HWP];
        float mean = 0.f;
        for (int j = 0; j < HWP; ++j) { v[j] = cb[(size_t)t * HWP + j]; mean += v[j]; }
        mean /= (float)HWP;
        float var = 0.f;
        for (int j = 0; j < HWP; ++j) { float d = v[j] - mean; var += d * d; }
        var /= (float)(HWP - 1);
        float inv = 1.f / sqrtf(var + 1e-5f);
        float mx = -3.4e38f;
        for (int j = 0; j < HWP; ++j) { v[j] = (v[j] - mean) * inv; mx = fmaxf(mx, v[j]); }
        float se = 0.f;
        for (int j = 0; j < HWP; ++j) { v[j] = expf(v[j] - mx); se += v[j]; }
        float rs = 1.f / se;
        for (int j = 0; j < HWP; ++j) sm[t * HWP + j] = v[j] * rs;
    }
    __syncthreads();
    if (t < HWP) {
        float s = 0.f;
        for (int i = 0; i < HWP; ++i) s += sm[i * HWP + t];
        attn_q[b * HWP + t] = s;
    }
}

// ----- attention pooling: mean/max over hw of attn[b,p]*x[b,c,p] -----
__global__ void k_pool(const float* __restrict__ x, const float* __restrict__ attn,
                       float* __restrict__ em, float* __restrict__ ee, int off) {
    int idx = blockIdx.x * blockDim.x + threadIdx.x;
    if (idx >= BB * CC) return;
    int b = idx / CC, c = idx % CC;
    const float* xb = x + ((size_t)b * CC + c) * HWP;
    const float* ab = attn + b * HWP;
    float mean = 0.f, mx = -3.4e38f;
    for (int p = 0; p < HWP; ++p) {
        float v = ab[p] * xb[p];
        mean += v;
        mx = fmaxf(mx, v);
    }
    em[(size_t)b * 1920 + off + c] = mean / (float)HWP;
    ee[(size_t)b * 1920 + off + c] = mx;
}

// ----- generic WMMA FC: out[M,N] = act(A[M,K] * W[N,K]^T + bias). -----
// grid (N/64, ceil(M/16)); block 128 = 4 waves along N. K % 64 == 0, N % 64 == 0.
#define ACT_NONE 0
#define ACT_RELU 1
#define ACT_HSW  2
__global__ void k_fc(const float* __restrict__ A, const float* __restrict__ Wt,
                     const float* __restrict__ bias, float* __restrict__ out,
                     int M, int N, int K, int act) {
    int wave = threadIdx.x >> 5, lane = threadIdx.x & 31;
    int nt = blockIdx.x * 4 + wave;
    int mt = blockIdx.y;
    int m = lane & 15, kk = (lane >> 4) << 1;
    int arow = mt * 16 + m;
    int wrow = nt * 16 + m;
    float maskA = (arow < M) ? 1.f : 0.f;
    const float* Ar = A + (size_t)((arow < M) ? arow : 0) * K;
    const float* Wr = Wt + (size_t)wrow * K;
    v8f acc = {};
    for (int k0 = 0; k0 < K; k0 += 64) {
        __builtin_prefetch(Wr + k0 + 64, 0, 3);
        for (int kq = 0; kq < 64; kq += 4) {
            int k = k0 + kq + kk;
            float a0 = maskA * Ar[k];
            float a1 = maskA * Ar[k + 1];
            float b0 = Wr[k];
            float b1 = Wr[k + 1];
            acc = wmma4(acc, a0, a1, b0, b1);
        }
    }
    int col = nt * 16 + (lane & 15);
    int rbase = mt * 16 + ((lane >> 4) << 3);
    for (int j = 0; j < 8; ++j) {
        int r = rbase + j;
        if (r < M) {
            float v = acc[j];
            if (bias) v += bias[col];
            if (act == ACT_RELU) v = fmaxf(v, 0.f);
            else if (act == ACT_HSW) v = v * fminf(fmaxf(v + 3.f, 0.f), 6.f) * (1.f / 6.f);
            out[(size_t)r * N + col] = v;
        }
    }
}

// ----- 2.5D group pooling: [64,960] -> mean over groups of 8 -> [8,960] -----
__global__ void k_gpool(const float* __restrict__ em2, const float* __restrict__ ee2,
                        float* __restrict__ pm, float* __restrict__ pe) {
    int idx = blockIdx.x * blockDim.x + threadIdx.x;
    if (idx >= 2 * 8 * CC) return;
    int sel = idx / (8 * CC); int rem = idx % (8 * CC);
    int r = rem / CC, d = rem % CC;
    const float* src = sel ? ee2 : em2;
    float s = 0.f;
    for (int g = 0; g < 8; ++g) s += src[(size_t)(r * 8 + g) * CC + d];
    (sel ? pe : pm)[rem] = s * 0.125f;
}

// ----- final tiny classifier heads: out[8,4] = [om | oe] -----
__global__ void k_cls2(const float* __restrict__ hm, const float* __restrict__ he,
                       const float* __restrict__ mw2, const float* __restrict__ mb2,
                       const float* __restrict__ ew2, const float* __restrict__ eb2,
                       float* __restrict__ out) {
    int t = threadIdx.x;
    if (t >= 32) return;
    int branch = t >> 4; int r = (t & 15) >> 1; int o = t & 1;
    const float* h  = branch ? he  : hm;
    const float* w  = branch ? ew2 : mw2;
    const float* bb = branch ? eb2 : mb2;
    float s = bb[o];
    for (int k = 0; k < 1920; ++k) s += h[(size_t)r * 1920 + k] * w[(size_t)o * 1920 + k];
    out[r * 4 + branch * 2 + o] = s;
}

// ---------------------------------------------------------------------------
extern "C" void kernel_launch(void* const* d_in, const int* in_sizes, int n_in,
                              void* d_out, int out_size, void* d_ws, size_t ws_size,
                              hipStream_t stream) {
    const float* x_p      = (const float*)d_in[0];
    const float* x_i      = (const float*)d_in[1];
    const float* scr_w_in = (const float*)d_in[2];
    const float* scr_bn_in= (const float*)d_in[3];
    const float* scr_w1   = (const float*)d_in[4];
    const float* scr_bn1  = (const float*)d_in[5];
    const float* scr_w2   = (const float*)d_in[6];
    const float* scr_bn2  = (const float*)d_in[7];
    const float* scr_w_out= (const float*)d_in[8];
    const float* scr_bn_o = (const float*)d_in[9];
    const float* cca_w    = (const float*)d_in[10];
    const float* cca_bn   = (const float*)d_in[11];
    const float* c0c2_w   = (const float*)d_in[12];
    const float* c0c2_bn  = (const float*)d_in[13];
    const float* c0c1_w   = (const float*)d_in[14];
    const float* c0c1_bn  = (const float*)d_in[15];
    const float* c0p_w    = (const float*)d_in[16];
    const float* c0p_bn   = (const float*)d_in[17];
    const float* c1c2_w   = (const float*)d_in[18];
    const float* c1c2_bn  = (const float*)d_in[19];
    const float* c1c1_w   = (const float*)d_in[20];
    const float* c1c1_bn  = (const float*)d_in[21];
    const float* c1p_w    = (const float*)d_in[22];
    const float* c1p_bn   = (const float*)d_in[23];
    const float* meta_w1  = (const float*)d_in[24];
    const float* meta_w2  = (const float*)d_in[25];
    const float* meta_b2  = (const float*)d_in[26];
    const float* ene_w1   = (const float*)d_in[27];
    const float* ene_w2   = (const float*)d_in[28];
    const float* ene_b2   = (const float*)d_in[29];
    const float* mcls_w1  = (const float*)d_in[30];
    const float* mcls_b1  = (const float*)d_in[31];
    const float* mcls_w2  = (const float*)d_in[32];
    const float* mcls_b2  = (const float*)d_in[33];
    const float* ecls_w1  = (const float*)d_in[34];
    const float* ecls_b1  = (const float*)d_in[35];
    const float* ecls_w2  = (const float*)d_in[36];
    const float* ecls_b2  = (const float*)d_in[37];

    float* ws = (float*)d_ws;
    size_t off = 0;
    auto alloc = [&](size_t n) { float* p = ws + off; off += n; return p; };
    float* xn    = alloc((size_t)BB * CC * HWP);
    float* g1    = alloc((size_t)BB * CO * HWP * UV);
    float* t2    = alloc((size_t)BB * CO * HWP * 9);
    float* t3    = alloc((size_t)BB * CO * HWP);
    float* xpn   = alloc((size_t)BB * CC * HWP);
    float* xin   = alloc((size_t)BB * CC * HWP);
    float* spt   = alloc((size_t)BB * CO * HWP);
    float* qry   = alloc((size_t)BB * CO * HWP);
    float* corr  = alloc((size_t)BB * HWP * HWP);
    float* ta    = alloc((size_t)BB * HWP * HWP);
    float* tb    = alloc((size_t)BB * HWP * HWP);
    float* c4a   = alloc((size_t)BB * 16 * HWP * HWP);
    float* s1    = alloc((size_t)BB * 16 * HWP * HWP);
    float* s2    = alloc((size_t)BB * 16 * HWP * HWP);
    float* c4b   = alloc((size_t)BB * HWP * HWP);
    float* atn_s = alloc((size_t)BB * HWP);
    float* atn_q = alloc((size_t)BB * HWP);
    float* em    = alloc((size_t)BB * 1920);
    float* ee    = alloc((size_t)BB * 1920);
    float* h1    = alloc((size_t)BB * 1920);
    float* h2    = alloc((size_t)BB * 1920);
    float* em2   = alloc((size_t)BB * CC);
    float* ee2   = alloc((size_t)BB * CC);
    float* pm    = alloc((size_t)8 * CC);
    float* pe    = alloc((size_t)8 * CC);
    float* hm    = alloc((size_t)8 * 1920);
    float* he    = alloc((size_t)8 * 1920);
    (void)ws_size; (void)n_in; (void)in_sizes; (void)out_size;

    for (int branch = 0; branch < 2; ++branch) {
        const float* x = branch ? x_i : x_p;
        float* dst = branch ? xin : xpn;
        float* z   = branch ? qry : spt;
        k_prep<<<BB * HWP, 128, 0, stream>>>(x, xn);
        k_gemm1<<<dim3(CO / 16, BB * UV), 128, 0, stream>>>(xn, scr_w_in, scr_bn_in, g1);
        {
            int total = BB * CO * HWP * 9;
            k_conv1<<<(total + 255) / 256, 256, 0, stream>>>(g1, scr_w1, scr_bn1, t2);
        }
        {
            int total = BB * CO * HWP;
            k_conv2<<<(total + 255) / 256, 256, 0, stream>>>(t2, scr_w2, scr_bn2, t3);
        }
        k_gemm_out<<<dim3(CC / 16, BB), 128, 0, stream>>>(t3, scr_w_out, scr_bn_o, x, dst);
        k_meansub<<<BB * HWP, 128, 0, stream>>>(dst);
        k_gemm_cca<<<dim3(CO / 16, BB), 128, 0, stream>>>(dst, cca_w, cca_bn, z);
        k_l2n96<<<BB * HWP, 128, 0, stream>>>(z);
    }

    k_gemm_corr<<<dim3(4, BB), 128, 0, stream>>>(spt, qry, corr);

    {
        int total = BB * HWP * HWP;
        k_sep1a<<<(total + 255) / 256, 256, 0, stream>>>(corr, c0c2_w, c0c2_bn, ta);
        k_sep1b<<<(total + 255) / 256, 256, 0, stream>>>(ta, c0c1_w, c0c1_bn, tb);
    }
    {
        int total = BB * 16 * HWP * HWP;
        k_sep1c<<<(total + 255) / 256, 256, 0, stream>>>(tb, c0p_w, c0p_bn, c4a);
        k_sep2a<<<(total + 255) / 256, 256, 0, stream>>>(c4a, c1c2_w, c1c2_bn, s1);
        k_sep2b<<<(total + 255) / 256, 256, 0, stream>>>(s1, c1c1_w, c1c1_bn, s2);
    }
    {
        int total = BB * HWP * HWP;
        k_sep2c<<<(total + 255) / 256, 256, 0, stream>>>(s2, c1p_w, c1p_bn, c4b);
    }
    k_attn<<<BB, 64, 0, stream>>>(c4b, atn_s, atn_q);

    {
        int total = BB * CC;
        k_pool<<<(total + 255) / 256, 256, 0, stream>>>(xpn, atn_s, em, ee, 0);
        k_pool<<<(total + 255) / 256, 256, 0, stream>>>(xin, atn_q, em, ee, CC);
    }

    // meta / energy heads: [64,1920] -> relu -> [64,960] (+bias)
    k_fc<<<dim3(1920 / 64, BB / 16), 128, 0, stream>>>(em, meta_w1, nullptr, h1, BB, 1920, 1920, ACT_RELU);
    k_fc<<<dim3(CC / 64,   BB / 16), 128, 0, stream>>>(h1, meta_w2, meta_b2, em2, BB, CC, 1920, ACT_NONE);
    k_fc<<<dim3(1920 / 64, BB / 16), 128, 0, stream>>>(ee, ene_w1, nullptr, h2, BB, 1920, 1920, ACT_RELU);
    k_fc<<<dim3(CC / 64,   BB / 16), 128, 0, stream>>>(h2, ene_w2, ene_b2, ee2, BB, CC, 1920, ACT_NONE);

    {
        int total = 2 * 8 * CC;
        k_gpool<<<(total + 255) / 256, 256, 0, stream>>>(em2, ee2, pm, pe);
    }

    // classifier heads: [8,960] -> hardswish([8,1920]) -> [8,2] each
    k_fc<<<dim3(1920 / 64, 1), 128, 0, stream>>>(pm, mcls_w1, mcls_b1, hm, 8, 1920, CC, ACT_HSW);
    k_fc<<<dim3(1920 / 64, 1), 128, 0, stream>>>(pe, ecls_w1, ecls_b1, he, 8, 1920, CC, ACT_HSW);
    k_cls2<<<1, 32, 0, stream>>>(hm, he, mcls_w2, mcls_b2, ecls_w2, ecls_b2, (float*)d_out);
}